// Slicing_14499809591771
// MI455X (gfx1250) — compile-verified
//
#include <hip/hip_runtime.h>
#include <cmath>

typedef float v2f __attribute__((ext_vector_type(2)));
typedef float v8f __attribute__((ext_vector_type(8)));

#define GH 16
#define GW 16
#define GD 8
#define NC 12
#define HH 1024
#define WW 1024
#define NB 8
#define ROWS 16

// SWAPX16: lane L <-> lane L^16 (group-of-32 swizzle, xor=0x10, and=0x1f)
__device__ __forceinline__ float swap16(float v) {
    return __int_as_float(__builtin_amdgcn_ds_swizzle(__float_as_int(v), 0x401F));
}

__global__ __launch_bounds__(256) void hdrnet_slice_wmma(
    const float* __restrict__ grid,   // [8,12,8,16,16]  (B,C,D,Hg,Wg)
    const float* __restrict__ guide,  // [8,1,1024,1024]
    float* __restrict__ out)          // [8,12,1024,1024]
{
    const int lane = threadIdx.x & 31;
    const int wave = threadIdx.x >> 5;
    const int b    = blockIdx.z;
    const int x0   = blockIdx.x * 256 + wave * 32;   // 32-aligned x-tile per wave
    const int y0   = blockIdx.y * ROWS;              // 16-aligned row strip

    const int hi = lane >> 4;   // half-wave (K-half / pixel sub-block)
    const int ch = lane & 15;   // channel (N) index

    // ---------- x weights (fx uniform within any 32-aligned tile) ----------
    const int x = x0 + lane;
    float gx  = ((float)x + 0.5f) * (1.0f / 64.0f);
    float fxf = floorf(gx - 0.5f);
    float tx  = gx - 0.5f - fxf;
    float wx0 = 1.0f - tx, wx1 = tx;
    int ifx = (int)fxf;
    int xi0 = min(max(ifx,     0), GW - 1);
    int xi1 = min(max(ifx + 1, 0), GW - 1);

    // swapped-half copies (y-invariant -> hoisted out of the row loop)
    float wx0_sw = swap16(wx0);
    float wx1_sw = swap16(wx1);
    // per-group x weights: group g sources pixel data from lane-half g
    float wxg[2][2];
    wxg[0][0] = hi ? wx0_sw : wx0;
    wxg[0][1] = hi ? wx1_sw : wx1;
    wxg[1][0] = hi ? wx0    : wx0_sw;
    wxg[1][1] = hi ? wx1    : wx1_sw;

    // ---------- y cell (fy uniform within the 16-row block) ----------
    float gyc = ((float)y0 + 0.5f) * (1.0f / 64.0f);
    float fyf = floorf(gyc - 0.5f);
    int ify = (int)fyf;
    int yi0 = min(max(ify,     0), GH - 1);
    int yi1 = min(max(ify + 1, 0), GH - 1);

    // ---------- B matrices: 4 cells x 2 K-chunks x 2 VGPRs, loaded once ----------
    // B(k,n): vgpr = k&1, lane = n + 16*(k>>1); chunk q covers z = 4q..4q+3
    float bm[4][2][2];
#pragma unroll
    for (int cell = 0; cell < 4; ++cell) {
        int yi = (cell & 2) ? yi1 : yi0;
        int xi = (cell & 1) ? xi1 : xi0;
#pragma unroll
        for (int q = 0; q < 2; ++q)
#pragma unroll
        for (int j = 0; j < 2; ++j) {
            int z = 4 * q + 2 * hi + j;
            bm[cell][q][j] = (ch < NC)
                ? grid[(((size_t)(b * NC + ch) * GD + z) * GH + yi) * GW + xi]
                : 0.0f;
        }
    }

    const float* grow = guide + (size_t)b * HH * WW + (size_t)y0 * WW + x;
    float* obase = (ch < NC) ? (out + ((size_t)(b * NC + ch) * HH + y0) * WW)
                             : out;  // dummy, never stored

    for (int ry = 0; ry < ROWS; ++ry) {
        float gv = grow[(size_t)ry * WW];
        float gz = gv * 8.0f;

        // dense z weights; boundary clipping folds into clamped end weights
        float wz[8];
        wz[0] = fminf(fmaxf(1.5f - gz, 0.0f), 1.0f);
#pragma unroll
        for (int z = 1; z <= 6; ++z)
            wz[z] = fmaxf(1.0f - fabsf(((float)z + 0.5f) - gz), 0.0f);
        wz[7] = fminf(fmaxf(gz - 6.5f, 0.0f), 1.0f);

        float wz_sw[8];
#pragma unroll
        for (int z = 0; z < 8; ++z) wz_sw[z] = swap16(wz[z]);

        float gy  = ((float)(y0 + ry) + 0.5f) * (1.0f / 64.0f);
        float ty  = gy - 0.5f - fyf;
        float wy0 = 1.0f - ty, wy1 = ty;

#pragma unroll
        for (int g = 0; g < 2; ++g) {
            // lane needs z-weight index 4q + 2*hi + j of pixel (lane&15)+16g;
            // pixel data comes from own lane if hi==g, else from swapped copy
            float wzsel[2][2];
#pragma unroll
            for (int q = 0; q < 2; ++q)
#pragma unroll
            for (int j = 0; j < 2; ++j) {
                float lo = (g == 0) ? wz[4*q + j]        : wz_sw[4*q + j];
                float hv = (g == 0) ? wz_sw[4*q + 2 + j] : wz[4*q + 2 + j];
                wzsel[q][j] = hi ? hv : lo;
            }
            float wxy[4];
            wxy[0] = wy0 * wxg[g][0];
            wxy[1] = wy0 * wxg[g][1];
            wxy[2] = wy1 * wxg[g][0];
            wxy[3] = wy1 * wxg[g][1];

            v8f acc = {};
#pragma unroll
            for (int cell = 0; cell < 4; ++cell)
#pragma unroll
            for (int q = 0; q < 2; ++q) {
                v2f a, bb;
                a[0]  = wxy[cell] * wzsel[q][0];
                a[1]  = wxy[cell] * wzsel[q][1];
                bb[0] = bm[cell][q][0];
                bb[1] = bm[cell][q][1];
                acc = __builtin_amdgcn_wmma_f32_16x16x4_f32(
                        false, a, false, bb, (short)0, acc, false, false);
            }

            // D: lane-half hi holds pixels m = 8*hi + v (v=0..7) of channel ch
            if (ch < NC) {
                float* o = obase + (size_t)ry * WW + (x0 + g * 16 + hi * 8);
                float4 s0 = make_float4(acc[0], acc[1], acc[2], acc[3]);
                float4 s1 = make_float4(acc[4], acc[5], acc[6], acc[7]);
                *(float4*)o       = s0;
                *((float4*)o + 1) = s1;
            }
        }
    }
}

extern "C" void kernel_launch(void* const* d_in, const int* in_sizes, int n_in,
                              void* d_out, int out_size, void* d_ws, size_t ws_size,
                              hipStream_t stream) {
    (void)in_sizes; (void)n_in; (void)out_size; (void)d_ws; (void)ws_size;
    const float* grid  = (const float*)d_in[0];
    const float* guide = (const float*)d_in[1];
    float* out = (float*)d_out;
    dim3 block(256, 1, 1);
    dim3 g(WW / 256, HH / ROWS, NB);   // (4, 64, 8)
    hipLaunchKernelGGL(hdrnet_slice_wmma, g, block, 0, stream, grid, guide, out);
}